// RegionProposalNetwork_70884140253474
// MI455X (gfx1250) — compile-verified
//
#include <hip/hip_runtime.h>
#include <stdint.h>

// ---------------------------------------------------------------------------
// RPN proposal generation for MI455X (gfx1250, wave32).
// Pipeline: flatten obj -> 2-level radix-threshold top-k -> decode/clip ->
// per-image bitonic sort (async-to-LDS staged) -> WMMA-tiled IoU suppression
// bit-matrix -> sequential greedy scan -> top-1000 output.
// Workspace layout (~57 MB) documented below; assumes ws_size is sufficient.
// ---------------------------------------------------------------------------

#define N_IMG   16
#define TOT     159882          // total anchors across levels
#define K_CAND  4507            // 1000+1000+1000+1000+507
#define WORDS   141             // ceil(K_CAND/32)
#define NPAD    8192            // sort padding (pow2 >= K_CAND)
#define NT      282             // ceil(K_CAND/16) tiles
#define NEG_INF (-1e30f)
#define XCLIP   4.135166556742356f   // log(1000/16)

typedef float v2f __attribute__((ext_vector_type(2)));
typedef float v8f __attribute__((ext_vector_type(8)));

#if defined(__HIP_DEVICE_COMPILE__) && __has_builtin(__builtin_amdgcn_wmma_f32_16x16x4_f32)
#define HAVE_WMMA_F32 1
#endif
#if defined(__HIP_DEVICE_COMPILE__) && __has_builtin(__builtin_amdgcn_global_load_async_to_lds_b64)
#define HAVE_ASYNC_LDS 1
#endif

// ---- workspace offsets (bytes) --------------------------------------------
static constexpr size_t SZ_OBJ = (size_t)N_IMG * TOT * 4;          // 10,232,448
static constexpr size_t SZ_H   = (size_t)N_IMG * 5 * 4096 * 4;     //  1,310,720
static constexpr size_t SZ_CS  = (size_t)N_IMG * K_CAND * 4;       //    288,448
static constexpr size_t OFF_OBJ = 0;
static constexpr size_t OFF_H1  = OFF_OBJ + SZ_OBJ;
static constexpr size_t OFF_H2  = OFF_H1 + SZ_H;
static constexpr size_t OFF_THR = OFF_H2 + SZ_H;                   // 80*4 ints
static constexpr size_t OFF_CNT = OFF_THR + 80 * 4 * 4;            // 160 ints
static constexpr size_t OFF_CSC = OFF_CNT + 160 * 4;
static constexpr size_t OFF_CIX = OFF_CSC + SZ_CS;
static constexpr size_t OFF_BOX = OFF_CIX + SZ_CS;                 // K*4 floats
static constexpr size_t OFF_SSC = OFF_BOX + SZ_CS * 4;
static constexpr size_t OFF_SBX = OFF_SSC + SZ_CS;
static constexpr size_t OFF_SLV = OFF_SBX + SZ_CS * 4;
static constexpr size_t OFF_MAT = OFF_SLV + SZ_CS;                 // N*K*WORDS u32

// ---- helpers --------------------------------------------------------------
__device__ __forceinline__ int level_of(int f) {
  if (f < 120000) return 0;
  if (f < 150000) return 1;
  if (f < 157500) return 2;
  if (f < 159375) return 3;
  return 4;
}
__device__ __forceinline__ int lvl_off(int l) {
  switch (l) { case 0: return 0; case 1: return 120000; case 2: return 150000;
               case 3: return 157500; default: return 159375; }
}
__device__ __forceinline__ void lvl_dims(int l, int& H, int& W) {
  switch (l) { case 0: H = W = 200; break; case 1: H = W = 100; break;
               case 2: H = W = 50; break; case 3: H = W = 25; break;
               default: H = W = 13; }
}
__device__ __forceinline__ int lvl_ksel(int l) { return l == 4 ? 507 : 1000; }
__device__ __forceinline__ int lvl_coff(int l) { return l * 1000; }

__device__ __forceinline__ unsigned fkey(float f) {
  unsigned u = __float_as_uint(f);
  return u ^ ((u & 0x80000000u) ? 0xFFFFFFFFu : 0x80000000u);
}

__device__ __forceinline__ unsigned ballot32(bool p) {
#if defined(__HIP_DEVICE_COMPILE__) && __has_builtin(__builtin_amdgcn_ballot_w32)
  return __builtin_amdgcn_ballot_w32(p);
#else
  return (unsigned)__ballot(p);
#endif
}

// ---- kernels --------------------------------------------------------------
__global__ void k_zero_u32(unsigned* p, long long n) {
  long long i = (long long)blockIdx.x * blockDim.x + threadIdx.x;
  long long st = (long long)gridDim.x * blockDim.x;
  for (; i < n; i += st) p[i] = 0u;
}

__global__ void k_flatten(const float* __restrict__ c0, const float* __restrict__ c1,
                          const float* __restrict__ c2, const float* __restrict__ c3,
                          const float* __restrict__ c4, float* __restrict__ obj) {
  long long t = (long long)blockIdx.x * blockDim.x + threadIdx.x;
  long long st = (long long)gridDim.x * blockDim.x;
  const long long NE = (long long)N_IMG * TOT;
  for (; t < NE; t += st) {
    int n = (int)(t / TOT), f = (int)(t % TOT);
    int l = level_of(f);
    int j = f - lvl_off(l);
    int cell = j / 3, a = j - cell * 3;
    int H, W; lvl_dims(l, H, W);
    int y = cell / W, x = cell - y * W;
    const float* c = (l == 0 ? c0 : l == 1 ? c1 : l == 2 ? c2 : l == 3 ? c3 : c4);
    obj[t] = c[(((size_t)n * 3 + a) * H + y) * (size_t)W + x];
  }
}

__global__ void k_hist1(const float* __restrict__ obj, unsigned* __restrict__ h1) {
  long long t = (long long)blockIdx.x * blockDim.x + threadIdx.x;
  long long st = (long long)gridDim.x * blockDim.x;
  const long long NE = (long long)N_IMG * TOT;
  for (; t < NE; t += st) {
    __builtin_prefetch(obj + t + st, 0, 0);   // -> global_prefetch_b8
    int n = (int)(t / TOT), f = (int)(t % TOT), l = level_of(f);
    unsigned u = fkey(obj[t]);
    atomicAdd(&h1[((size_t)n * 5 + l) * 4096 + (u >> 20)], 1u);
  }
}

__global__ void k_thr1(const unsigned* __restrict__ h1, int* __restrict__ thr) {
  int b = blockIdx.x;            // (n,l) pair, 0..79
  if (threadIdx.x != 0) return;
  int l = b % 5, k = lvl_ksel(l);
  const unsigned* h = h1 + (size_t)b * 4096;
  int cum = 0, B = 0, need = k;
  for (int bin = 4095; bin >= 0; --bin) {
    int c = (int)h[bin]; cum += c;
    if (cum >= k) { B = bin; need = k - (cum - c); break; }
  }
  thr[b * 4 + 0] = B; thr[b * 4 + 1] = need;
}

__global__ void k_hist2(const float* __restrict__ obj, const int* __restrict__ thr,
                        unsigned* __restrict__ h2) {
  long long t = (long long)blockIdx.x * blockDim.x + threadIdx.x;
  long long st = (long long)gridDim.x * blockDim.x;
  const long long NE = (long long)N_IMG * TOT;
  for (; t < NE; t += st) {
    __builtin_prefetch(obj + t + st, 0, 0);
    int n = (int)(t / TOT), f = (int)(t % TOT), l = level_of(f);
    unsigned u = fkey(obj[t]);
    if ((int)(u >> 20) == thr[(n * 5 + l) * 4])
      atomicAdd(&h2[((size_t)n * 5 + l) * 4096 + ((u >> 8) & 0xFFFu)], 1u);
  }
}

__global__ void k_thr2(const unsigned* __restrict__ h2, int* __restrict__ thr) {
  int b = blockIdx.x;
  if (threadIdx.x != 0) return;
  int need1 = thr[b * 4 + 1];
  const unsigned* h = h2 + (size_t)b * 4096;
  int cum = 0, B = -1, need = need1;
  for (int bin = 4095; bin >= 0; --bin) {
    int c = (int)h[bin]; cum += c;
    if (cum >= need1) { B = bin; need = need1 - (cum - c); break; }
  }
  thr[b * 4 + 2] = B; thr[b * 4 + 3] = need;
}

__global__ void k_compact(const float* __restrict__ obj, const int* __restrict__ thr,
                          int* __restrict__ cnt, float* __restrict__ csc,
                          int* __restrict__ cix) {
  long long t = (long long)blockIdx.x * blockDim.x + threadIdx.x;
  long long st = (long long)gridDim.x * blockDim.x;
  const long long NE = (long long)N_IMG * TOT;
  for (; t < NE; t += st) {
    int n = (int)(t / TOT), f = (int)(t % TOT), l = level_of(f);
    float sc = obj[t];
    unsigned u = fkey(sc);
    int id4 = (n * 5 + l) * 4;
    int B1 = thr[id4 + 0], B2 = thr[id4 + 2], need2 = thr[id4 + 3];
    int b1 = (int)(u >> 20);
    bool take = false;
    if (b1 > B1) take = true;
    else if (b1 == B1) {
      int b2 = (int)((u >> 8) & 0xFFFu);
      if (b2 > B2) take = true;
      else if (b2 == B2) {
        int tp = atomicAdd(&cnt[80 + n * 5 + l], 1);
        if (tp < need2) take = true;
      }
    }
    if (take) {
      int p = atomicAdd(&cnt[n * 5 + l], 1);
      if (p < lvl_ksel(l)) {
        long long dst = (long long)n * K_CAND + lvl_coff(l) + p;
        csc[dst] = sc; cix[dst] = f;
      }
    }
  }
}

__global__ void k_decode(const int* __restrict__ cix, const float* __restrict__ anchors,
                         const float* __restrict__ r0, const float* __restrict__ r1,
                         const float* __restrict__ r2, const float* __restrict__ r3,
                         const float* __restrict__ r4,
                         float* __restrict__ csc, float* __restrict__ box) {
  long long t = (long long)blockIdx.x * blockDim.x + threadIdx.x;
  if (t >= (long long)N_IMG * K_CAND) return;
  int n = (int)(t / K_CAND);
  int idx = cix[t];
  int l = level_of(idx);
  int j = idx - lvl_off(l);
  int cell = j / 3, a = j - cell * 3;
  int H, W; lvl_dims(l, H, W);
  int y = cell / W, x = cell - y * W;
  const float* an = anchors + (size_t)idx * 4;
  float a0 = an[0], a1 = an[1], a2 = an[2], a3 = an[3];
  float wa = a2 - a0, ha = a3 - a1;
  float cxa = a0 + 0.5f * wa, cya = a1 + 0.5f * ha;
  const float* rg = (l == 0 ? r0 : l == 1 ? r1 : l == 2 ? r2 : l == 3 ? r3 : r4);
  size_t hw = (size_t)H * W;
  size_t rb = (((size_t)n * 12 + a * 4) * H + y) * (size_t)W + x;
  float dx = rg[rb], dy = rg[rb + hw];
  float dw = fminf(rg[rb + 2 * hw], XCLIP);
  float dh = fminf(rg[rb + 3 * hw], XCLIP);
  float cx = dx * wa + cxa, cy = dy * ha + cya;
  float w = expf(dw) * wa, h = expf(dh) * ha;
  float x1 = fminf(fmaxf(cx - 0.5f * w, 0.f), 800.f);
  float y1 = fminf(fmaxf(cy - 0.5f * h, 0.f), 800.f);
  float x2 = fminf(fmaxf(cx + 0.5f * w, 0.f), 800.f);
  float y2 = fminf(fmaxf(cy + 0.5f * h, 0.f), 800.f);
  bool valid = (x2 - x1 >= 1e-3f) && (y2 - y1 >= 1e-3f);
  if (!valid) csc[t] = NEG_INF;
  box[t * 4 + 0] = x1; box[t * 4 + 1] = y1; box[t * 4 + 2] = x2; box[t * 4 + 3] = y2;
}

// per-image bitonic sort (descending score, stable on candidate slot)
__global__ __launch_bounds__(1024) void k_sort(const float* __restrict__ csc,
                                               const int* __restrict__ cix,
                                               const float* __restrict__ box,
                                               float* __restrict__ ssc,
                                               float* __restrict__ sbx,
                                               int* __restrict__ slv) {
  extern __shared__ unsigned char s_dyn[];
  uint64_t* keys = (uint64_t*)s_dyn;
  float* stage = (float*)(s_dyn + (size_t)NPAD * 8);
  int img = blockIdx.x, tid = threadIdx.x;
  long long base = (long long)img * K_CAND;

#if defined(HAVE_ASYNC_LDS)
  {  // CDNA5 async global->LDS staging of the K score words (ASYNCcnt path)
    typedef int v2i __attribute__((ext_vector_type(2)));
    typedef __attribute__((address_space(1))) v2i g_v2i;   // global int2
    typedef __attribute__((address_space(3))) v2i s_v2i;   // LDS int2
    const char* gsrc = (const char*)(csc + base);
    char* lbase = (char*)stage;
    int pairs = (K_CAND * 4 + 7) / 8;   // b64 transfers
    for (int i = tid; i < pairs; i += blockDim.x)
      __builtin_amdgcn_global_load_async_to_lds_b64(
          (g_v2i*)(const void*)(gsrc + (size_t)i * 8),
          (s_v2i*)(void*)(lbase + (size_t)i * 8), 0, 0);
#if __has_builtin(__builtin_amdgcn_s_wait_asynccnt)
    __builtin_amdgcn_s_wait_asynccnt(0);
#else
    asm volatile("s_wait_asynccnt 0x0" ::: "memory");
#endif
  }
#else
  for (int t = tid; t < K_CAND; t += blockDim.x) stage[t] = csc[base + t];
#endif
  __syncthreads();

  for (int t = tid; t < NPAD; t += blockDim.x) {
    uint64_t kk = 0;
    if (t < K_CAND)
      kk = ((uint64_t)fkey(stage[t]) << 32) | (uint64_t)(0xFFFFFFFFu - (unsigned)t);
    keys[t] = kk;
  }
  __syncthreads();

  for (unsigned size = 2; size <= NPAD; size <<= 1) {
    for (unsigned stride = size >> 1; stride > 0; stride >>= 1) {
      for (unsigned t = tid; t < NPAD / 2; t += blockDim.x) {
        unsigned i = 2 * t - (t & (stride - 1));
        unsigned jj = i + stride;
        bool desc = ((i & size) == 0);
        uint64_t A = keys[i], B = keys[jj];
        bool sw = desc ? (A < B) : (A > B);
        if (sw) { keys[i] = B; keys[jj] = A; }
      }
      __syncthreads();
    }
  }

  for (int t = tid; t < K_CAND; t += blockDim.x) {
    uint64_t kk = keys[t];
    unsigned slot = 0xFFFFFFFFu - (unsigned)(kk & 0xFFFFFFFFull);
    long long src = base + slot;
    ssc[base + t] = csc[src];
    sbx[(base + t) * 4 + 0] = box[src * 4 + 0];
    sbx[(base + t) * 4 + 1] = box[src * 4 + 1];
    sbx[(base + t) * 4 + 2] = box[src * 4 + 2];
    sbx[(base + t) * 4 + 3] = box[src * 4 + 3];
    slv[base + t] = level_of(cix[src]);
  }
}

// WMMA-tiled IoU suppression bit-matrix: one wave32 per 16x16 tile.
__global__ __launch_bounds__(32) void k_mat(const float* __restrict__ sbx,
                                            const int* __restrict__ slv,
                                            unsigned* __restrict__ mat) {
  int img = blockIdx.y;
  int rt = blockIdx.x / NT, ct = blockIdx.x % NT;
  if (ct < rt) return;                 // only j > i can be suppressed
  int l = threadIdx.x, ls = l & 15;
  bool lo = (l < 16);
  long long base = (long long)img * K_CAND;

  int rg = rt * 16 + ls; int rc = rg < K_CAND ? rg : K_CAND - 1;
  int cg = ct * 16 + ls; int cc = cg < K_CAND ? cg : K_CAND - 1;
  float4 rb = ((const float4*)sbx)[base + rc];
  float4 cb = ((const float4*)sbx)[base + cc];
  float roff = 1600.0f * (float)slv[base + rc];
  float coff = 1600.0f * (float)slv[base + cc];
  float rx1 = rb.x + roff, ry1 = rb.y + roff, rx2 = rb.z + roff, ry2 = rb.w + roff;
  float qx1 = cb.x + coff, qy1 = cb.y + coff, qx2 = cb.z + coff, qy2 = cb.w + coff;
  float rw = rx2 - rx1, rh = ry2 - ry1, ra = rw * rh;
  float cw = qx2 - qx1, ch = qy2 - qy1, ca = cw * ch;

#if defined(HAVE_WMMA_F32)
  // Outer sums/diffs via rank-2 matmuls: A row i = [a_i, 1, 0, 0],
  // B col j = [1, b_j, 0, 0]^T  =>  D[i][j] = a_i + b_j (neg b_j for diffs).
  float aone = lo ? 1.f : 0.f, bone = lo ? 1.f : 0.f;
  v8f z = {};
  v2f aA, bB;
  aA = (v2f){lo ? rw : 0.f, aone};  bB = (v2f){bone, lo ? cw : 0.f};
  v8f SW = __builtin_amdgcn_wmma_f32_16x16x4_f32(false, aA, false, bB, (short)0, z, false, false);
  aA = (v2f){lo ? rh : 0.f, aone};  bB = (v2f){bone, lo ? ch : 0.f};
  v8f SH = __builtin_amdgcn_wmma_f32_16x16x4_f32(false, aA, false, bB, (short)0, z, false, false);
  aA = (v2f){lo ? ra : 0.f, aone};  bB = (v2f){bone, lo ? ca : 0.f};
  v8f SA = __builtin_amdgcn_wmma_f32_16x16x4_f32(false, aA, false, bB, (short)0, z, false, false);
  aA = (v2f){lo ? rx1 : 0.f, aone}; bB = (v2f){bone, lo ? -qx1 : 0.f};
  v8f D1 = __builtin_amdgcn_wmma_f32_16x16x4_f32(false, aA, false, bB, (short)0, z, false, false);
  aA = (v2f){lo ? rx2 : 0.f, aone}; bB = (v2f){bone, lo ? -qx2 : 0.f};
  v8f D2 = __builtin_amdgcn_wmma_f32_16x16x4_f32(false, aA, false, bB, (short)0, z, false, false);
  aA = (v2f){lo ? ry1 : 0.f, aone}; bB = (v2f){bone, lo ? -qy1 : 0.f};
  v8f E1 = __builtin_amdgcn_wmma_f32_16x16x4_f32(false, aA, false, bB, (short)0, z, false, false);
  aA = (v2f){lo ? ry2 : 0.f, aone}; bB = (v2f){bone, lo ? -qy2 : 0.f};
  v8f E2 = __builtin_amdgcn_wmma_f32_16x16x4_f32(false, aA, false, bB, (short)0, z, false, false);
#endif

#pragma unroll
  for (int r = 0; r < 8; ++r) {
    int M = r + (lo ? 0 : 8);
    float sw_, sh_, sa_, d1_, d2_, e1_, e2_;
#if defined(HAVE_WMMA_F32)
    sw_ = SW[r]; sh_ = SH[r]; sa_ = SA[r];
    d1_ = D1[r]; d2_ = D2[r]; e1_ = E1[r]; e2_ = E2[r];
#else
    sw_ = __shfl(rw, M) + cw;  sh_ = __shfl(rh, M) + ch;  sa_ = __shfl(ra, M) + ca;
    d1_ = __shfl(rx1, M) - qx1; d2_ = __shfl(rx2, M) - qx2;
    e1_ = __shfl(ry1, M) - qy1; e2_ = __shfl(ry2, M) - qy2;
#endif
    float iw = 0.5f * (sw_ - fabsf(d2_) - fabsf(d1_)); iw = iw > 0.f ? iw : 0.f;
    float ih = 0.5f * (sh_ - fabsf(e2_) - fabsf(e1_)); ih = ih > 0.f ? ih : 0.f;
    float inter = iw * ih;
    // union = sa_ - inter + 1e-12 > 0 always, so iou > 0.7  <=>  inter > 0.7*union
    // (avoids the IEEE v_div_scale/v_rcp/v_div_fixup sequence per element)
    bool pred = (inter > 0.7f * (sa_ - inter + 1e-12f));
    int rowg = rt * 16 + M;
    int colg = ct * 16 + ls;
    pred = pred && (colg > rowg) && (colg < K_CAND);
    unsigned bal = ballot32(pred);
    if (l == 0) {
      int row = rt * 16 + r;
      unsigned m = bal & 0xFFFFu;
      if (row < K_CAND && m)
        atomicOr(&mat[((size_t)base + row) * WORDS + (ct >> 1)], m << ((ct & 1) * 16));
    }
    if (l == 1) {
      int row = rt * 16 + r + 8;
      unsigned m = bal >> 16;
      if (row < K_CAND && m)
        atomicOr(&mat[((size_t)base + row) * WORDS + (ct >> 1)], m << ((ct & 1) * 16));
    }
  }
}

// per-image sequential greedy scan + top-1000 emission
__global__ __launch_bounds__(256) void k_scan_out(const float* __restrict__ ssc,
                                                  const float* __restrict__ sbx,
                                                  const unsigned* __restrict__ mat,
                                                  float* __restrict__ out) {
  int img = blockIdx.x, tid = threadIdx.x;
  __shared__ unsigned remv[WORDS];
  long long base = (long long)img * K_CAND;
  for (int w = tid; w < WORDS; w += blockDim.x) remv[w] = 0u;
  __syncthreads();
  for (int i = tid; i < K_CAND; i += blockDim.x)
    if (ssc[base + i] <= NEG_INF * 0.5f) atomicOr(&remv[i >> 5], 1u << (i & 31));
  __syncthreads();
  for (int i = 0; i < K_CAND; ++i) {
    bool rem = (remv[i >> 5] >> (i & 31)) & 1u;
    __syncthreads();
    if (!rem) {
      const unsigned* row = mat + ((size_t)base + i) * WORDS;
      if (tid == 0) __builtin_prefetch(row + WORDS, 0, 0);
      for (int w = tid; w < WORDS; w += blockDim.x) remv[w] |= row[w];
    }
    __syncthreads();
  }
  float* o = out + (size_t)img * 5000;
  for (int t = tid; t < 5000; t += blockDim.x) o[t] = 0.f;
  __syncthreads();
  if (tid == 0) {
    int p = 0;
    for (int i = 0; i < K_CAND && p < 1000; ++i) {
      if (!((remv[i >> 5] >> (i & 31)) & 1u)) {
        o[p * 5 + 0] = sbx[(base + i) * 4 + 0];
        o[p * 5 + 1] = sbx[(base + i) * 4 + 1];
        o[p * 5 + 2] = sbx[(base + i) * 4 + 2];
        o[p * 5 + 3] = sbx[(base + i) * 4 + 3];
        o[p * 5 + 4] = 1.f / (1.f + expf(-ssc[base + i]));
        ++p;
      }
    }
  }
}

// ---------------------------------------------------------------------------
extern "C" void kernel_launch(void* const* d_in, const int* in_sizes, int n_in,
                              void* d_out, int out_size, void* d_ws, size_t ws_size,
                              hipStream_t stream) {
  const float* cls[5]; const float* reg[5];
  for (int i = 0; i < 5; ++i) {
    cls[i] = (const float*)d_in[2 * i];
    reg[i] = (const float*)d_in[2 * i + 1];
  }
  const float* anchors = (const float*)d_in[10];

  char* ws = (char*)d_ws;
  float*    obj = (float*)(ws + OFF_OBJ);
  unsigned* h1  = (unsigned*)(ws + OFF_H1);
  unsigned* h2  = (unsigned*)(ws + OFF_H2);
  int*      thr = (int*)(ws + OFF_THR);
  int*      cnt = (int*)(ws + OFF_CNT);
  float*    csc = (float*)(ws + OFF_CSC);
  int*      cix = (int*)(ws + OFF_CIX);
  float*    box = (float*)(ws + OFF_BOX);
  float*    ssc = (float*)(ws + OFF_SSC);
  float*    sbx = (float*)(ws + OFF_SBX);
  int*      slv = (int*)(ws + OFF_SLV);
  unsigned* mat = (unsigned*)(ws + OFF_MAT);
  float*    out = (float*)d_out;

  // zero mutable scratch every call (deterministic replay)
  k_zero_u32<<<2048, 256, 0, stream>>>(h1, (long long)N_IMG * 5 * 4096 * 2);
  k_zero_u32<<<1, 256, 0, stream>>>((unsigned*)cnt, 160);
  k_zero_u32<<<2048, 256, 0, stream>>>((unsigned*)csc, (long long)N_IMG * K_CAND * 2);
  k_zero_u32<<<4096, 256, 0, stream>>>(mat, (long long)N_IMG * K_CAND * WORDS);

  k_flatten<<<4096, 256, 0, stream>>>(cls[0], cls[1], cls[2], cls[3], cls[4], obj);
  k_hist1<<<4096, 256, 0, stream>>>(obj, h1);
  k_thr1<<<80, 32, 0, stream>>>(h1, thr);
  k_hist2<<<4096, 256, 0, stream>>>(obj, thr, h2);
  k_thr2<<<80, 32, 0, stream>>>(h2, thr);
  k_compact<<<4096, 256, 0, stream>>>(obj, thr, cnt, csc, cix);

  long long nd = (long long)N_IMG * K_CAND;
  k_decode<<<(unsigned)((nd + 255) / 256), 256, 0, stream>>>(
      cix, anchors, reg[0], reg[1], reg[2], reg[3], reg[4], csc, box);

  size_t smem = (size_t)NPAD * 8 + 4512 * 4;   // keys + score staging
  k_sort<<<N_IMG, 1024, smem, stream>>>(csc, cix, box, ssc, sbx, slv);

  dim3 gm(NT * NT, N_IMG);
  k_mat<<<gm, 32, 0, stream>>>(sbx, slv, mat);

  k_scan_out<<<N_IMG, 256, 0, stream>>>(ssc, sbx, mat, out);
}